// EncoderGENConv_80015240725025
// MI455X (gfx1250) — compile-verified
//
#include <hip/hip_runtime.h>
#include <hip/hip_bf16.h>

#define D_FEAT  128
#define EPS_MSG 1e-7f
#define EPS_SM  1e-16f

typedef float v2f __attribute__((ext_vector_type(2)));
typedef float v8f __attribute__((ext_vector_type(8)));
typedef int   v4i __attribute__((vector_size(4 * sizeof(int))));

#define AS1 __attribute__((address_space(1)))
#define AS3 __attribute__((address_space(3)))

#if __has_builtin(__builtin_amdgcn_global_load_async_to_lds_b128) && \
    __has_builtin(__builtin_amdgcn_s_wait_asynccnt)
#define HAVE_ASYNC_LDS 1
#else
#define HAVE_ASYNC_LDS 0
#endif

// 16-byte global -> LDS copy: async DMA (ASYNCcnt) when available.
__device__ __forceinline__ void copy16_g2l(const float* g, float* l) {
#if HAVE_ASYNC_LDS
    const AS1 void* gv = (const AS1 void*)g;   // addrspace cast (probe-verified OK)
    AS3 void*       lv = (AS3 void*)l;
    __builtin_amdgcn_global_load_async_to_lds_b128(
        (AS1 v4i*)gv, (AS3 v4i*)lv, 0, 0);
#else
    *(float4*)l = *(const float4*)g;
#endif
}

__device__ __forceinline__ void async_wait0() {
#if HAVE_ASYNC_LDS
    __builtin_amdgcn_s_wait_asynccnt(0);
#endif
}

// ---------------------------------------------------------------------------
// Edge pass 1: per-destination feature-wise max of m = relu(h[src]) + eps.
// m > 0 always, so float max == integer max on the raw bits (atomicMax u32).
// One wave (32 lanes) per edge, 4 floats per lane (float4 gather).
// ---------------------------------------------------------------------------
__global__ void edge_max_kernel(const float* __restrict__ h,
                                const int* __restrict__ ei,
                                unsigned* __restrict__ mx, int E) {
    int t = blockIdx.x * blockDim.x + threadIdx.x;
    int e = t >> 5;
    if (e >= E) return;
    int q   = (t & 31) << 2;
    int src = ei[e];
    int dst = ei[E + e];
    const float4 v = *(const float4*)(h + (size_t)src * D_FEAT + q);
    unsigned* p = mx + (size_t)dst * D_FEAT + q;
    atomicMax(p + 0, __float_as_uint(fmaxf(v.x, 0.f) + EPS_MSG));
    atomicMax(p + 1, __float_as_uint(fmaxf(v.y, 0.f) + EPS_MSG));
    atomicMax(p + 2, __float_as_uint(fmaxf(v.z, 0.f) + EPS_MSG));
    atomicMax(p + 3, __float_as_uint(fmaxf(v.w, 0.f) + EPS_MSG));
}

// ---------------------------------------------------------------------------
// Edge pass 2: den += exp(m - mx[dst]); num += exp(m - mx[dst]) * m.
// (aggr = num / (den + 1e-16) afterwards; alpha never materialized.)
// ---------------------------------------------------------------------------
__global__ void edge_sum_kernel(const float* __restrict__ h,
                                const int* __restrict__ ei,
                                const unsigned* __restrict__ mx,
                                float* __restrict__ den,
                                float* __restrict__ num, int E) {
    int t = blockIdx.x * blockDim.x + threadIdx.x;
    int e = t >> 5;
    if (e >= E) return;
    int q   = (t & 31) << 2;
    int src = ei[e];
    int dst = ei[E + e];
    const float4 v = *(const float4*)(h + (size_t)src * D_FEAT + q);
    size_t base = (size_t)dst * D_FEAT + q;
    float m0 = fmaxf(v.x, 0.f) + EPS_MSG;
    float m1 = fmaxf(v.y, 0.f) + EPS_MSG;
    float m2 = fmaxf(v.z, 0.f) + EPS_MSG;
    float m3 = fmaxf(v.w, 0.f) + EPS_MSG;
    float e0 = __expf(m0 - __uint_as_float(mx[base + 0]));
    float e1 = __expf(m1 - __uint_as_float(mx[base + 1]));
    float e2 = __expf(m2 - __uint_as_float(mx[base + 2]));
    float e3 = __expf(m3 - __uint_as_float(mx[base + 3]));
    atomicAdd(den + base + 0, e0);
    atomicAdd(den + base + 1, e1);
    atomicAdd(den + base + 2, e2);
    atomicAdd(den + base + 3, e3);
    atomicAdd(num + base + 0, e0 * m0);
    atomicAdd(num + base + 1, e1 * m1);
    atomicAdd(num + base + 2, e2 * m2);
    atomicAdd(num + base + 3, e3 * m3);
}

// ---------------------------------------------------------------------------
// Node update: h2 = h + num / (den + 1e-16)   (float4 vectorized)
// ---------------------------------------------------------------------------
__global__ void node_update_kernel(const float* __restrict__ hcur,
                                   const float* __restrict__ den,
                                   const float* __restrict__ num,
                                   float* __restrict__ h2, int n4) {
    int t = blockIdx.x * blockDim.x + threadIdx.x;
    if (t >= n4) return;
    float4 hv = ((const float4*)hcur)[t];
    float4 dv = ((const float4*)den)[t];
    float4 nv = ((const float4*)num)[t];
    float4 o;
    o.x = hv.x + nv.x / (dv.x + EPS_SM);
    o.y = hv.y + nv.y / (dv.y + EPS_SM);
    o.z = hv.z + nv.z / (dv.z + EPS_SM);
    o.w = hv.w + nv.w / (dv.w + EPS_SM);
    ((float4*)h2)[t] = o;
}

// ---------------------------------------------------------------------------
// fp32 WMMA GEMM: Y[N,OUT] = act(X[N,IN] @ W[IN,OUT] + bias)
// V_WMMA_F32_16X16X4_F32; A tile staged in padded LDS; per-coltile weight
// slab (IN x 16) staged in LDS via async DMA, double-buffered when it fits
// in a conservative 64KB budget. A frag (ISA 7.12.2): lanes 0-15 -> K{0,1},
// lanes 16-31 -> K{2,3}. C/D: VGPR i -> row i + 8*(lane>=16), col = lane&15.
// ---------------------------------------------------------------------------
template <int IN, int OUT, int WAVES, bool RELU>
__global__ void gemm_wmma_kernel(const float* __restrict__ X,
                                 const float* __restrict__ W,
                                 const float* __restrict__ bias,
                                 float* __restrict__ Y, int nrows) {
    extern __shared__ float lds[];
    constexpr int PAD   = 4;
    constexpr int LDSTR = IN + PAD;
    constexpr int ROWS  = WAVES * 16;
    constexpr int NT    = WAVES * 32;
    constexpr int ASIZE = ROWS * LDSTR;
    constexpr int WSLAB = IN * 16;
    constexpr bool DB   = (ASIZE + 2 * WSLAB) * 4 <= 65536;  // double-buffer?
    constexpr int CT    = OUT / 16;

    float* atile = lds;
    float* wbuf0 = lds + ASIZE;
    float* wbuf1 = DB ? (wbuf0 + WSLAB) : wbuf0;

    const int r0  = blockIdx.x * ROWS;
    const int tid = threadIdx.x;

    // Stage weight slab for col-tile c0 into dst (row k, col c layout, stride 16).
    auto stage_w = [&](float* dst, int c0) {
        for (int idx = tid * 4; idx < WSLAB; idx += NT * 4) {
            int row = idx >> 4;        // k
            int q   = idx & 15;        // col offset (multiple of 4)
            copy16_g2l(W + (size_t)row * OUT + c0 + q, dst + idx);
        }
    };

    // Stage the block's A tile (row-clamped so every lane has a valid addr).
    for (int idx = tid * 4; idx < ROWS * IN; idx += NT * 4) {
        int row = idx / IN;
        int k   = idx % IN;
        int g   = r0 + row;
        if (g >= nrows) g = nrows - 1;
        copy16_g2l(X + (size_t)g * IN + k, atile + row * LDSTR + k);
    }
    stage_w(wbuf0, 0);
    async_wait0();
    __syncthreads();

    const int wave  = tid >> 5;
    const int lane  = tid & 31;
    const int half  = lane >> 4;     // 0: lanes 0-15, 1: lanes 16-31
    const int lr    = lane & 15;
    const int kbase = half * 2;      // K pair selected by lane half
    const float* arow = atile + (wave * 16 + lr) * LDSTR;

    for (int ct = 0; ct < CT; ++ct) {
        float* wcur = (DB && (ct & 1)) ? wbuf1 : wbuf0;
        if (DB) {  // prefetch next slab while computing this one
            float* wnext = (ct & 1) ? wbuf0 : wbuf1;
            if (ct + 1 < CT) stage_w(wnext, (ct + 1) * 16);
        }

        const int c0  = ct * 16;
        const int col = c0 + lr;
        v8f acc;
        const float bv = bias[col];
#pragma unroll
        for (int i = 0; i < 8; ++i) acc[i] = bv;

#pragma unroll
        for (int kb = 0; kb < IN; kb += 4) {
            v2f a = *(const v2f*)(arow + kb + kbase);      // A[r][kb+kbase..+1]
            v2f b;
            b.x = wcur[(kb + kbase) * 16 + lr];            // B[kb+kbase][col]
            b.y = wcur[(kb + kbase + 1) * 16 + lr];        // B[kb+kbase+1][col]
            acc = __builtin_amdgcn_wmma_f32_16x16x4_f32(
                false, a, false, b, (short)0, acc, false, false);
        }

#pragma unroll
        for (int i = 0; i < 8; ++i) {
            int row = r0 + wave * 16 + half * 8 + i;
            if (row < nrows) {
                float o = acc[i];
                if (RELU) o = fmaxf(o, 0.f);
                Y[(size_t)row * OUT + col] = o;
            }
        }

        if (DB) {
            async_wait0();      // prefetch into wnext landed
            __syncthreads();    // everyone done reading wcur
        } else {
            __syncthreads();    // all reads of wbuf0 done
            if (ct + 1 < CT) {
                stage_w(wbuf0, (ct + 1) * 16);
                async_wait0();
                __syncthreads();
            }
        }
    }
}

// ---------------------------------------------------------------------------
// Driver. Input order (jax tree_flatten, dict keys sorted):
//   0: x [N,128]    1: edge_index [2,E] (int32)
//   2+4l: conv{l}.W1   3+4l: conv{l}.W2   4+4l: conv{l}.b1   5+4l: conv{l}.b2
//   14+2l: lin{l}.W    15+2l: lin{l}.b
// ---------------------------------------------------------------------------
extern "C" void kernel_launch(void* const* d_in, const int* in_sizes, int n_in,
                              void* d_out, int out_size, void* d_ws, size_t ws_size,
                              hipStream_t stream) {
    const float* x  = (const float*)d_in[0];
    const int*   ei = (const int*)d_in[1];
    const int N = in_sizes[0] / D_FEAT;
    const int E = in_sizes[1] / 2;

    const size_t nd = (size_t)N * D_FEAT;
    float* MX  = (float*)d_ws;      // used as u32 max bits
    float* DEN = MX  + nd;
    float* NUM = DEN + nd;
    float* H2  = NUM + nd;
    float* T1  = H2  + nd;          // [N,256]
    float* T2  = T1  + 2 * nd;      // [N,128]
    float* HN  = T2  + nd;          // hidden state between layers

    // LDS bytes, mirroring the kernel's constexpr layout.
    auto shbytes = [](int IN, int WAVES) -> size_t {
        int asize = WAVES * 16 * (IN + 4);
        int wslab = IN * 16;
        bool db = (asize + 2 * wslab) * 4 <= 65536;
        return (size_t)(asize + (db ? 2 : 1) * wslab) * sizeof(float);
    };

    const int ethreads = E * 32;
    const int egrid    = (ethreads + 255) / 256;
    const int n4       = (int)(nd / 4);
    const int ngrid    = (n4 + 255) / 256;

    for (int l = 0; l < 3; ++l) {
        const float* hcur = (l == 0) ? x : HN;
        float*       hdst = (l == 2) ? (float*)d_out : HN;
        const float* W1 = (const float*)d_in[2 + 4 * l];
        const float* W2 = (const float*)d_in[3 + 4 * l];
        const float* b1 = (const float*)d_in[4 + 4 * l];
        const float* b2 = (const float*)d_in[5 + 4 * l];
        const float* LW = (const float*)d_in[14 + 2 * l];
        const float* Lb = (const float*)d_in[15 + 2 * l];

        (void)hipMemsetAsync(MX, 0, 3 * nd * sizeof(float), stream);
        edge_max_kernel<<<egrid, 256, 0, stream>>>(hcur, ei, (unsigned*)MX, E);
        edge_sum_kernel<<<egrid, 256, 0, stream>>>(hcur, ei, (const unsigned*)MX,
                                                   DEN, NUM, E);
        node_update_kernel<<<ngrid, 256, 0, stream>>>(hcur, DEN, NUM, H2, n4);

        {   // t1 = relu(h2 @ W1 + b1)   [N,128]x[128,256]
            int grid = (N + 63) / 64;
            gemm_wmma_kernel<128, 256, 4, true>
                <<<grid, 128, shbytes(128, 4), stream>>>(H2, W1, b1, T1, N);
        }
        {   // t2 = t1 @ W2 + b2         [N,256]x[256,128]
            int grid = (N + 31) / 32;
            gemm_wmma_kernel<256, 128, 2, false>
                <<<grid, 64, shbytes(256, 2), stream>>>(T1, W2, b2, T2, N);
        }
        {   // h = relu(t2 @ linW + linb) [N,128]x[128,128]
            int grid = (N + 63) / 64;
            gemm_wmma_kernel<128, 128, 4, true>
                <<<grid, 128, shbytes(128, 4), stream>>>(T2, LW, Lb, hdst, N);
        }
    }
}